// Model_58634893525678
// MI455X (gfx1250) — compile-verified
//
#include <hip/hip_runtime.h>
#include <hip/hip_bf16.h>

#define DD   128
#define GG   128
#define LL   4
#define NCC  10
#define EPSF 1e-5f

typedef float v2f __attribute__((ext_vector_type(2)));
typedef float v8f __attribute__((ext_vector_type(8)));

// ---------------------------------------------------------------------------
// GEMM: H[nrows x 128] = X[nrows x 128] @ W[128 x 128]  (f32 WMMA path)
// block = 256 threads (8 wave32); block covers 16 rows; wave w owns cols 16w..16w+15
// ---------------------------------------------------------------------------
__global__ void __launch_bounds__(256)
gemm_wmma_f32(const float* __restrict__ X, const float* __restrict__ W,
              float* __restrict__ H, int nrows) {
  __shared__ float xs[16 * 129];              // pitch 129 -> conflict-free column reads
  const int row0 = blockIdx.x * 16;
  const int tid  = threadIdx.x;

  for (int i = tid; i < 16 * 128; i += 256) {
    int r = i >> 7, c = i & 127;
    int gr = row0 + r;
    xs[r * 129 + c] = (gr < nrows) ? X[(size_t)gr * DD + c] : 0.f;
  }
  __syncthreads();

  const int wave  = tid >> 5;                 // 0..7
  const int lane  = tid & 31;
  const int lh    = lane >> 4;                // lane half: 0 or 1
  const int l     = lane & 15;
  const int col0  = wave * 16;

  v8f acc = {};
#pragma unroll
  for (int k0 = 0; k0 < 128; k0 += 4) {
    // A 16x4 f32: lanes 0-15 -> K = k0+0/1 ; lanes 16-31 -> K = k0+2/3 (M = l)
    v2f a, b;
    a.x = xs[l * 129 + k0 + 2 * lh + 0];
    a.y = xs[l * 129 + k0 + 2 * lh + 1];
    // B 4x16 f32 (mirror layout): row K, column N = col0 + l
    b.x = W[(size_t)(k0 + 2 * lh + 0) * DD + col0 + l];
    b.y = W[(size_t)(k0 + 2 * lh + 1) * DD + col0 + l];
    acc = __builtin_amdgcn_wmma_f32_16x16x4_f32(false, a, false, b,
                                                (short)0, acc, false, false);
  }

  // C/D layout: VGPR j, lanes 0-15 -> M=j, N=lane ; lanes 16-31 -> M=j+8
  const int m_off = lh * 8;
#pragma unroll
  for (int j = 0; j < 8; ++j) {
    int gr = row0 + m_off + j;
    if (gr < nrows) H[(size_t)gr * DD + col0 + l] = acc[j];
  }
}

// ---------------------------------------------------------------------------
// Degree / normalization helpers
// ---------------------------------------------------------------------------
__global__ void deg_init(float* deg, int n) {
  int i = blockIdx.x * blockDim.x + threadIdx.x;
  if (i < n) deg[i] = 1.0f;                   // self-loop
}
__global__ void deg_accum(const int* __restrict__ dst, float* deg, int e) {
  int i = blockIdx.x * blockDim.x + threadIdx.x;
  if (i < e) atomicAdd(&deg[dst[i]], 1.0f);
}
__global__ void rsqrt_inplace(float* deg, int n) {
  int i = blockIdx.x * blockDim.x + threadIdx.x;
  if (i < n) deg[i] = rsqrtf(deg[i]);         // deg -> d^-1/2
}

// agg[i,:] = h[i,:] * dis[i]^2   (self-loop contribution; also initializes agg)
__global__ void selfloop_init(const float* __restrict__ H, const float* __restrict__ dis,
                              float* __restrict__ AGG, int n) {
  long long idx = (long long)blockIdx.x * blockDim.x + threadIdx.x;
  if (idx < (long long)n * DD) {
    int node = (int)(idx >> 7);
    float d = dis[node];
    AGG[idx] = H[idx] * d * d;
  }
}

// one wave per edge: lane handles 4 features (float4 load + 4 atomic f32 adds)
__global__ void __launch_bounds__(256)
edge_scatter(const float* __restrict__ H, const int* __restrict__ src,
             const int* __restrict__ dst, const float* __restrict__ dis,
             float* __restrict__ AGG, int e) {
  long long t = (long long)blockIdx.x * blockDim.x + threadIdx.x;
  int edge = (int)(t >> 5);
  int lane = (int)(t & 31);
  if (edge >= e) return;
  int s = src[edge], d = dst[edge];
  float w = dis[s] * dis[d];
  const float4* hp = reinterpret_cast<const float4*>(H + (size_t)s * DD) + lane;
  float4 v = *hp;
  float* ap = AGG + (size_t)d * DD + lane * 4;
  atomicAdd(ap + 0, v.x * w);
  atomicAdd(ap + 1, v.y * w);
  atomicAdd(ap + 2, v.z * w);
  atomicAdd(ap + 3, v.w * w);
}

// ---------------------------------------------------------------------------
// Per-graph reductions (batch sorted -> contiguous ranges, binary search)
// ---------------------------------------------------------------------------
__device__ __forceinline__ int lower_bound_i(const int* a, int n, int key) {
  int lo = 0, hi = n;
  while (lo < hi) { int mid = (lo + hi) >> 1; if (a[mid] < key) lo = mid + 1; else hi = mid; }
  return lo;
}

__global__ void __launch_bounds__(256)
graph_mean(const float* __restrict__ AGG, const int* __restrict__ batch,
           float* __restrict__ mean, int n) {
  int g = blockIdx.x;
  int s = lower_bound_i(batch, n, g);
  int e = lower_bound_i(batch, n, g + 1);
  int t = threadIdx.x, f = t & 127, p = t >> 7;
  float sum = 0.f;
  for (int i = s + p; i < e; i += 2) sum += AGG[(size_t)i * DD + f];
  __shared__ float red[256];
  red[t] = sum; __syncthreads();
  if (t < 128) {
    float cnt = (float)((e - s) > 0 ? (e - s) : 1);
    mean[g * DD + t] = (red[t] + red[t + 128]) / cnt;
  }
}

__global__ void __launch_bounds__(256)
graph_var(const float* __restrict__ AGG, const int* __restrict__ batch,
          const float* __restrict__ mean, const float* __restrict__ cb,
          const float* __restrict__ gms, float* __restrict__ var, int n) {
  int g = blockIdx.x;
  int s = lower_bound_i(batch, n, g);
  int e = lower_bound_i(batch, n, g + 1);
  int t = threadIdx.x, f = t & 127, p = t >> 7;
  float mh = mean[g * DD + f] + cb[f];        // mean of h = mean(agg) + bias
  float ms = gms[f];
  float sum = 0.f;
  for (int i = s + p; i < e; i += 2) {
    float sub = (AGG[(size_t)i * DD + f] + cb[f]) - ms * mh;
    sum += sub * sub;
  }
  __shared__ float red[256];
  red[t] = sum; __syncthreads();
  if (t < 128) {
    float cnt = (float)((e - s) > 0 ? (e - s) : 1);
    var[g * DD + t] = (red[t] + red[t + 128]) / cnt;
  }
}

// GraphNorm affine + ELU + residual; optionally mirror to d_out (last layer)
__global__ void layer_finalize(const float* __restrict__ AGG, const int* __restrict__ batch,
                               const float* __restrict__ mean, const float* __restrict__ var,
                               const float* __restrict__ cb, const float* __restrict__ gw,
                               const float* __restrict__ gb, const float* __restrict__ gms,
                               float* __restrict__ X, float* __restrict__ OUT2, int n) {
  long long idx = (long long)blockIdx.x * blockDim.x + threadIdx.x;
  if (idx >= (long long)n * DD) return;
  int node = (int)(idx >> 7), f = (int)(idx & 127);
  int g = batch[node];
  float mh  = mean[g * DD + f] + cb[f];
  float sub = (AGG[idx] + cb[f]) - gms[f] * mh;
  float y   = gw[f] * sub * rsqrtf(var[g * DD + f] + EPSF) + gb[f];
  y = (y > 0.f) ? y : expm1f(y);              // ELU (alpha = 1)
  float xn = y + X[idx];                      // residual
  X[idx] = xn;
  if (OUT2) OUT2[idx] = xn;
}

__global__ void __launch_bounds__(256)
graph_pool(const float* __restrict__ X, const int* __restrict__ batch,
           float* __restrict__ pooled, int n) {
  int g = blockIdx.x;
  int s = lower_bound_i(batch, n, g);
  int e = lower_bound_i(batch, n, g + 1);
  int t = threadIdx.x, f = t & 127, p = t >> 7;
  float sum = 0.f;
  for (int i = s + p; i < e; i += 2) sum += X[(size_t)i * DD + f];
  __shared__ float red[256];
  red[t] = sum; __syncthreads();
  if (t < 128) pooled[g * DD + t] = red[t] + red[t + 128];
}

__global__ void bn_relu(float* __restrict__ Z1, const float* __restrict__ b1,
                        const float* __restrict__ gamma, const float* __restrict__ beta,
                        const float* __restrict__ bnm, const float* __restrict__ bnv) {
  int idx = blockIdx.x * blockDim.x + threadIdx.x;
  if (idx >= GG * DD) return;
  int f = idx & 127;
  float z = Z1[idx] + b1[f];
  z = gamma[f] * (z - bnm[f]) * rsqrtf(bnv[f] + EPSF) + beta[f];
  Z1[idx] = fmaxf(z, 0.f);
}

__global__ void head_out(const float* __restrict__ Z1, const float* __restrict__ W2,
                         const float* __restrict__ b2, float* __restrict__ ZOUT) {
  int idx = blockIdx.x * blockDim.x + threadIdx.x;
  if (idx >= GG * NCC) return;
  int g = idx / NCC, c = idx % NCC;
  float acc = b2[c];
#pragma unroll 8
  for (int k = 0; k < DD; ++k) acc += Z1[g * DD + k] * W2[k * NCC + c];
  ZOUT[idx] = acc;
}

// ---------------------------------------------------------------------------
extern "C" void kernel_launch(void* const* d_in, const int* in_sizes, int n_in,
                              void* d_out, int out_size, void* d_ws, size_t ws_size,
                              hipStream_t stream) {
  const float* x_in = (const float*)d_in[0];
  const int*   ei   = (const int*)d_in[1];
  const int*   batch= (const int*)d_in[2];
  const float* cw   = (const float*)d_in[3];
  const float* cb   = (const float*)d_in[4];
  const float* gw   = (const float*)d_in[5];
  const float* gb   = (const float*)d_in[6];
  const float* gms  = (const float*)d_in[7];
  const float* w1   = (const float*)d_in[8];
  const float* b1   = (const float*)d_in[9];
  const float* bng  = (const float*)d_in[10];
  const float* bnb  = (const float*)d_in[11];
  const float* bnm  = (const float*)d_in[12];
  const float* bnv  = (const float*)d_in[13];
  const float* w2   = (const float*)d_in[14];
  const float* b2   = (const float*)d_in[15];

  const int N = in_sizes[2];                  // nodes (batch has one entry per node)
  const int E = in_sizes[1] / 2;              // edges
  const int* src = ei;
  const int* dst = ei + E;

  float* out = (float*)d_out;                 // [ N*D node feats | G*NC logits ]
  float* outX = out;
  float* outZ = out + (size_t)N * DD;

  // workspace layout
  float* Xb   = (float*)d_ws;                 // N*D
  float* Hb   = Xb + (size_t)N * DD;          // N*D
  float* Ab   = Hb + (size_t)N * DD;          // N*D
  float* dis  = Ab + (size_t)N * DD;          // N (deg -> d^-1/2)
  float* mean = dis + N;                      // G*D
  float* var  = mean + GG * DD;               // G*D
  float* pooled = var + GG * DD;              // G*D
  float* z1   = pooled + GG * DD;             // G*D

  const long long ND = (long long)N * DD;
  const int bElem = (int)((ND + 255) / 256);
  const int bN    = (N + 255) / 256;
  const int bE    = (E + 255) / 256;
  const int bScat = (int)(((long long)E * 32 + 255) / 256);
  const int bRows = (N + 15) / 16;

  // x working copy (inputs must stay untouched)
  hipMemcpyAsync(Xb, x_in, (size_t)ND * sizeof(float),
                 hipMemcpyDeviceToDevice, stream);

  // symmetric normalization d^-1/2
  deg_init<<<bN, 256, 0, stream>>>(dis, N);
  deg_accum<<<bE, 256, 0, stream>>>(dst, dis, E);
  rsqrt_inplace<<<bN, 256, 0, stream>>>(dis, N);

  for (int i = 0; i < LL; ++i) {
    const float* Wi  = cw  + (size_t)i * DD * DD;
    const float* cbi = cb  + (size_t)i * DD;
    const float* gwi = gw  + (size_t)i * DD;
    const float* gbi = gb  + (size_t)i * DD;
    const float* gmi = gms + (size_t)i * DD;

    gemm_wmma_f32<<<bRows, 256, 0, stream>>>(Xb, Wi, Hb, N);
    selfloop_init<<<bElem, 256, 0, stream>>>(Hb, dis, Ab, N);
    edge_scatter<<<bScat, 256, 0, stream>>>(Hb, src, dst, dis, Ab, E);
    graph_mean<<<GG, 256, 0, stream>>>(Ab, batch, mean, N);
    graph_var<<<GG, 256, 0, stream>>>(Ab, batch, mean, cbi, gmi, var, N);
    layer_finalize<<<bElem, 256, 0, stream>>>(Ab, batch, mean, var, cbi, gwi, gbi, gmi,
                                              Xb, (i == LL - 1) ? outX : nullptr, N);
  }

  // head: global add pool -> dense(WMMA) -> BN+ReLU -> dense
  graph_pool<<<GG, 256, 0, stream>>>(Xb, batch, pooled, N);
  gemm_wmma_f32<<<GG / 16, 256, 0, stream>>>(pooled, w1, z1, GG);
  bn_relu<<<(GG * DD + 255) / 256, 256, 0, stream>>>(z1, b1, bng, bnb, bnm, bnv);
  head_out<<<(GG * NCC + 255) / 256, 256, 0, stream>>>(z1, w2, b2, outZ);
}